// conv_1x1_9_layer_19052474925377
// MI455X (gfx1250) — compile-verified
//
#include <hip/hip_runtime.h>
#include <hip/hip_bf16.h>

typedef float v2f __attribute__((ext_vector_type(2)));
typedef float v8f __attribute__((ext_vector_type(8)));

#define CIN   64
#define COUT  64
#define HW    65536           // 256*256
#define WLDSTRIDE 68          // padded row stride (floats): conflict-free ds_load_b64

// ---------------------------------------------------------------------------
// Prep: Wavg[o][c] = mean_n weights[n][o][c] ; bavg[o] = mean_n biases[n][o]
// ws layout: [0,4096) = Wavg row-major [Cout][Cin], [4096,4160) = bavg
// ---------------------------------------------------------------------------
__global__ __launch_bounds__(256) void avg_weights_kernel(
    const float* __restrict__ w,     // [9][64][64]
    const float* __restrict__ bias,  // [9][64]
    float* __restrict__ ws)
{
    int idx = blockIdx.x * 256 + threadIdx.x;
    if (idx < COUT * CIN) {
        float s = 0.f;
#pragma unroll
        for (int n = 0; n < 9; ++n) s += w[n * COUT * CIN + idx];
        ws[idx] = s * (1.0f / 9.0f);
    }
    if (idx < COUT) {
        float s = 0.f;
#pragma unroll
        for (int n = 0; n < 9; ++n) s += bias[n * COUT + idx];
        ws[COUT * CIN + idx] = s * (1.0f / 9.0f);
    }
}

// ---------------------------------------------------------------------------
// Main: Y = Wavg * X + b  via V_WMMA_F32_16X16X4_F32
// One wave -> 16 contiguous pixels x all 64 Couts (4 C tiles of 16x16).
// ---------------------------------------------------------------------------
__global__ __launch_bounds__(256) void conv1x1_wmma_kernel(
    const float* __restrict__ x,     // [16][64][65536]
    const float* __restrict__ ws,    // Wavg + bavg
    float* __restrict__ out)         // [16][64][65536]
{
    __shared__ float lds_w[COUT * WLDSTRIDE];
    __shared__ float lds_b[COUT];

    const int t = threadIdx.x;

    // Stage averaged weights into LDS with padded stride (bank-conflict-free A loads)
#pragma unroll
    for (int i = 0; i < (COUT * CIN) / 256; ++i) {
        int idx = t + i * 256;
        int o = idx >> 6, c = idx & 63;
        lds_w[o * WLDSTRIDE + c] = ws[idx];
    }
    if (t < COUT) lds_b[t] = ws[COUT * CIN + t];
    __syncthreads();

    const int lane = t & 31;           // wave32
    const int wv   = t >> 5;           // 8 waves / block
    const int grp  = blockIdx.x * 8 + wv;      // 16-pixel group id, 65536 total
    const int P0   = grp << 4;                 // global pixel index
    const int bb   = P0 >> 16;                 // batch  (P0 / HW)
    const int p    = P0 & (HW - 1);            // pixel within image

    const float* xc = x   + (size_t)bb * CIN  * HW + p + (lane & 15);
    float*       oc = out + (size_t)bb * COUT * HW + p + (lane & 15);

    const int mrow = lane & 15;                 // M index within 16-row tile
    const int koff = (lane >> 4) << 1;          // lanes 0-15 -> K{0,1}; 16-31 -> K{2,3}

    v8f acc0 = {}, acc1 = {}, acc2 = {}, acc3 = {};

#pragma unroll
    for (int k0 = 0; k0 < CIN; k0 += 4) {
        // B fragment: X[k0+koff .. +1][p + lane%16]
        v2f bf;
        bf.x = xc[(size_t)(k0 + koff)     * HW];
        bf.y = xc[(size_t)(k0 + koff + 1) * HW];

        // A fragments: Wavg rows, contiguous K pair -> ds_load_b64
        const int kc = k0 + koff;
        v2f a0 = *(const v2f*)&lds_w[(mrow     ) * WLDSTRIDE + kc];
        v2f a1 = *(const v2f*)&lds_w[(mrow + 16) * WLDSTRIDE + kc];
        v2f a2 = *(const v2f*)&lds_w[(mrow + 32) * WLDSTRIDE + kc];
        v2f a3 = *(const v2f*)&lds_w[(mrow + 48) * WLDSTRIDE + kc];

        acc0 = __builtin_amdgcn_wmma_f32_16x16x4_f32(false, a0, false, bf, (short)0, acc0, false, false);
        acc1 = __builtin_amdgcn_wmma_f32_16x16x4_f32(false, a1, false, bf, (short)0, acc1, false, false);
        acc2 = __builtin_amdgcn_wmma_f32_16x16x4_f32(false, a2, false, bf, (short)0, acc2, false, false);
        acc3 = __builtin_amdgcn_wmma_f32_16x16x4_f32(false, a3, false, bf, (short)0, acc3, false, false);
    }

    // Bias as rank-1 update: A = bavg column (K=0 only), B = ones row (K=0 only)
    {
        v2f bo; bo.x = (lane < 16) ? 1.0f : 0.0f; bo.y = 0.0f;
        v2f ab; ab.y = 0.0f;
        ab.x = (lane < 16) ? lds_b[mrow +  0] : 0.0f;
        acc0 = __builtin_amdgcn_wmma_f32_16x16x4_f32(false, ab, false, bo, (short)0, acc0, false, false);
        ab.x = (lane < 16) ? lds_b[mrow + 16] : 0.0f;
        acc1 = __builtin_amdgcn_wmma_f32_16x16x4_f32(false, ab, false, bo, (short)0, acc1, false, false);
        ab.x = (lane < 16) ? lds_b[mrow + 32] : 0.0f;
        acc2 = __builtin_amdgcn_wmma_f32_16x16x4_f32(false, ab, false, bo, (short)0, acc2, false, false);
        ab.x = (lane < 16) ? lds_b[mrow + 48] : 0.0f;
        acc3 = __builtin_amdgcn_wmma_f32_16x16x4_f32(false, ab, false, bo, (short)0, acc3, false, false);
    }

    // Store: C/D tile layout -> VGPR r, lanes 0-15: M=r, lanes 16-31: M=r+8
    const int rbase = (lane >> 4) << 3;
#pragma unroll
    for (int r = 0; r < 8; ++r) {
        oc[(size_t)( 0 + rbase + r) * HW] = acc0[r];
        oc[(size_t)(16 + rbase + r) * HW] = acc1[r];
        oc[(size_t)(32 + rbase + r) * HW] = acc2[r];
        oc[(size_t)(48 + rbase + r) * HW] = acc3[r];
    }
}

extern "C" void kernel_launch(void* const* d_in, const int* in_sizes, int n_in,
                              void* d_out, int out_size, void* d_ws, size_t ws_size,
                              hipStream_t stream)
{
    const float* x       = (const float*)d_in[0];  // [16][64][256][256]
    const float* weights = (const float*)d_in[1];  // [9][64][64][1][1]
    const float* biases  = (const float*)d_in[2];  // [9][64]
    float*       out     = (float*)d_out;
    float*       ws      = (float*)d_ws;           // needs 4160 floats

    avg_weights_kernel<<<16, 256, 0, stream>>>(weights, biases, ws);

    // 2^20 pixels / (16 per wave * 8 waves per block) = 8192 blocks
    conv1x1_wmma_kernel<<<8192, 256, 0, stream>>>(x, ws, out);
}